// MambaTextEncoder_6038724018451
// MI455X (gfx1250) — compile-verified
//
#include <hip/hip_runtime.h>
#include <hip/hip_bf16.h>

typedef float v2f __attribute__((ext_vector_type(2)));
typedef float v8f __attribute__((ext_vector_type(8)));

#define BDIM   8
#define LDIM   1024
#define DDIM   128
#define IDIM   256
#define NDIM   16
#define KCONV  4
#define DTRDIM 8
#define MROWS  (BDIM*LDIM)   /* 8192 rows (b,l) */
#define SSMLD  48            /* ssm padded 40 -> 48 cols */

__device__ __forceinline__ float siluf(float v) { return v / (1.0f + expf(-v)); }

/* ---------------- embedding gather: x[m,d] = embed[ids[m],d] ---------------- */
__global__ void k_embed(const int* __restrict__ ids, const float* __restrict__ embed,
                        float* __restrict__ x) {
  int idx = blockIdx.x * blockDim.x + threadIdx.x;   // MROWS*DDIM
  int m = idx >> 7, d = idx & 127;
  x[idx] = embed[ids[m] * DDIM + d];
}

/* ---------------- rmsnorm over D=128; one row per 128-thread block ---------- */
__global__ void k_rmsnorm(const float* __restrict__ x, const float* __restrict__ w,
                          float* __restrict__ out) {
  __shared__ float sp[4];
  int row = blockIdx.x, t = threadIdx.x;
  float v = x[row * DDIM + t];
  float s = v * v;
  for (int off = 16; off >= 1; off >>= 1) s += __shfl_xor(s, off);
  if ((t & 31) == 0) sp[t >> 5] = s;
  __syncthreads();
  float tot = sp[0] + sp[1] + sp[2] + sp[3];
  float scale = rsqrtf(tot * (1.0f / DDIM) + 1e-5f);
  out[row * DDIM + t] = v * scale * w[t];
}

/* ---------------- WMMA f32 GEMM: C(MxN) = A(MxK) * W^T, W is (N,K) row-major.
   One wave per 16x16 tile via V_WMMA_F32_16X16X4_F32.
   Fragment layout (ISA 7.12.2): A 16x4 f32: lane m=lane&15, vgpr0={K0|K2},
   vgpr1={K1|K3} split on lane>=16 -> per-lane contiguous pair A[m][k0+2*hf..+1].
   B 4x16: same K split, lanes hold N -> contiguous pair W[n][k0+2*hf..+1].
   C/D: vgpr e -> row e+8*hf, col lane&15.
   N-edge handling is branch-free: clamp row, always load, scale fragment by a
   per-lane 0/1 mask (keeps EXEC all-1s through the WMMA loop, no saveexec).
   MODE 0: store; MODE 1: store softplus(v+bias[n]); MODE 2: C += v (residual). */
template<int MODE>
__global__ __launch_bounds__(128)
void k_gemm(const float* __restrict__ A, const float* __restrict__ W,
            float* __restrict__ C, const float* __restrict__ bias,
            int K, int lda, int ldw, int ldc, int nTilesN, int Nreal) {
  int wave = threadIdx.x >> 5;
  int lane = threadIdx.x & 31;
  int tile = blockIdx.x * 4 + wave;
  int tm = tile / nTilesN, tn = tile % nTilesN;
  int m0 = tm * 16, n0 = tn * 16;
  int r = lane & 15, hf = lane >> 4;
  int wrow = n0 + r;
  float wmask = (wrow < Nreal) ? 1.0f : 0.0f;
  int wclamp = (wrow < Nreal) ? wrow : (Nreal - 1);
  const float* arow = A + (size_t)(m0 + r) * lda + 2 * hf;
  const float* wrp  = W + (size_t)wclamp * ldw + 2 * hf;
  v8f acc = {};
  for (int kb = 0; kb < K; kb += 4) {
    v2f a = *(const v2f*)(arow + kb);
    v2f b = *(const v2f*)(wrp + kb);
    b.x *= wmask;               // branch-free edge masking; co-executes with WMMA
    b.y *= wmask;
    acc = __builtin_amdgcn_wmma_f32_16x16x4_f32(false, a, false, b,
                                                (short)0, acc, false, false);
  }
  float* ccol = C + n0 + r;
  float bv = (MODE == 1) ? bias[n0 + r] : 0.0f;
#pragma unroll
  for (int e = 0; e < 8; ++e) {
    int row = m0 + e + 8 * hf;
    float v = acc[e];
    if (MODE == 1) { float z = v + bv; v = (z > 20.f) ? z : log1pf(expf(z)); }
    if (MODE == 2) ccol[(size_t)row * ldc] += v;
    else           ccol[(size_t)row * ldc]  = v;
  }
}

/* ---------------- causal depthwise conv K=4 + bias + SiLU --------------------
   u = proj[:, :, 0:I]; uc[b,l,i] = silu(sum_k w[i,k]*u[b,l-3+k,i] + cb[i]) */
__global__ void k_conv(const float* __restrict__ proj, const float* __restrict__ w,
                       const float* __restrict__ cb, float* __restrict__ uc) {
  int idx = blockIdx.x * blockDim.x + threadIdx.x;   // B*L*I
  int i = idx & (IDIM - 1);
  int bl = idx >> 8;            // b*L + l
  int l = bl & (LDIM - 1);
  float s = cb[i];
#pragma unroll
  for (int k = 0; k < KCONV; ++k) {
    int li = l - (KCONV - 1) + k;
    if (li >= 0)
      s += w[i * KCONV + k] * proj[(size_t)(bl - (KCONV - 1) + k) * (2 * IDIM) + i];
  }
  uc[idx] = siluf(s);
}

/* ---------------- selective scan --------------------------------------------
   One lane per (i,n); 16-lane groups reduce over n each step.
   Fuses dA = exp(dt*A), dBu, y = h.C, +uc*D, *silu(gate). */
__global__ __launch_bounds__(256)
void k_scan(const float* __restrict__ dt, const float* __restrict__ uc,
            const float* __restrict__ ssm, const float* __restrict__ proj,
            const float* __restrict__ A_log, const float* __restrict__ Dp,
            float* __restrict__ y) {
  int b = blockIdx.x >> 4;
  int iblk = blockIdx.x & 15;
  int t = threadIdx.x;
  int il = t >> 4, n = t & 15;
  int i = iblk * 16 + il;
  float Aneg = -expf(A_log[i * NDIM + n]);
  float Di = Dp[i];
  float h = 0.f;
  int mbase = b * LDIM;
  for (int l = 0; l < LDIM; ++l) {
    int m = mbase + l;
    if (l + 8 < LDIM) {           // hide L2 latency on the streamed rows
      __builtin_prefetch(&dt[(size_t)(m + 8) * IDIM + i], 0, 0);
      __builtin_prefetch(&ssm[(size_t)(m + 8) * SSMLD + n], 0, 0);
    }
    float dtv = dt[(size_t)m * IDIM + i];
    float ucv = uc[(size_t)m * IDIM + i];
    float Bn  = ssm[(size_t)m * SSMLD + DTRDIM + n];
    float Cn  = ssm[(size_t)m * SSMLD + DTRDIM + NDIM + n];
    float a = expf(dtv * Aneg);
    h = a * h + (dtv * ucv) * Bn;
    float p = h * Cn;
    p += __shfl_xor(p, 1);
    p += __shfl_xor(p, 2);
    p += __shfl_xor(p, 4);
    p += __shfl_xor(p, 8);
    if (n == 0) {
      float g = proj[(size_t)m * (2 * IDIM) + IDIM + i];
      y[(size_t)m * IDIM + i] = (p + ucv * Di) * siluf(g);
    }
  }
}

/* ---------------- final rmsnorm of x[:,0,:] -> out (8x128) ------------------ */
__global__ void k_finalnorm(const float* __restrict__ x, const float* __restrict__ w,
                            float* __restrict__ out) {
  __shared__ float sp[4];
  int bidx = blockIdx.x, t = threadIdx.x;
  size_t row = (size_t)bidx * LDIM;   // l = 0
  float v = x[row * DDIM + t];
  float s = v * v;
  for (int off = 16; off >= 1; off >>= 1) s += __shfl_xor(s, off);
  if ((t & 31) == 0) sp[t >> 5] = s;
  __syncthreads();
  float tot = sp[0] + sp[1] + sp[2] + sp[3];
  out[bidx * DDIM + t] = v * rsqrtf(tot * (1.0f / DDIM) + 1e-5f) * w[t];
}

extern "C" void kernel_launch(void* const* d_in, const int* in_sizes, int n_in,
                              void* d_out, int out_size, void* d_ws, size_t ws_size,
                              hipStream_t stream) {
  const int*   ids     = (const int*)  d_in[0];
  const float* embed   = (const float*)d_in[1];
  const float* norm_w  = (const float*)d_in[2];
  const float* in_w    = (const float*)d_in[3];
  const float* conv_w  = (const float*)d_in[4];
  const float* conv_b  = (const float*)d_in[5];
  const float* xp_w    = (const float*)d_in[6];
  const float* dt_w    = (const float*)d_in[7];
  const float* dt_b    = (const float*)d_in[8];
  const float* A_log   = (const float*)d_in[9];
  const float* Dp      = (const float*)d_in[10];
  const float* out_w   = (const float*)d_in[11];
  const float* normf_w = (const float*)d_in[12];
  float* out = (float*)d_out;

  float* ws   = (float*)d_ws;
  float* x    = ws;                       // 8192*128
  float* hn   = x    + MROWS * DDIM;      // 8192*128
  float* proj = hn   + MROWS * DDIM;      // 8192*512
  float* uc   = proj + MROWS * 2 * IDIM;  // 8192*256
  float* ssm  = uc   + MROWS * IDIM;      // 8192*48
  float* dtb  = ssm  + MROWS * SSMLD;     // 8192*256
  float* yb   = dtb  + MROWS * IDIM;      // 8192*256

  k_embed<<<MROWS * DDIM / 256, 256, 0, stream>>>(ids, embed, x);

  for (int l = 0; l < 4; ++l) {
    k_rmsnorm<<<MROWS, 128, 0, stream>>>(x, norm_w + l * DDIM, hn);

    /* proj = hn @ in_w^T : M=8192 K=128 N=512 -> 512*32 tiles */
    k_gemm<0><<<(MROWS / 16) * 32 / 4, 128, 0, stream>>>(
        hn, in_w + (size_t)l * 2 * IDIM * DDIM, proj, nullptr,
        DDIM, DDIM, DDIM, 2 * IDIM, 32, 2 * IDIM);

    k_conv<<<MROWS * IDIM / 256, 256, 0, stream>>>(
        proj, conv_w + (size_t)l * IDIM * KCONV, conv_b + (size_t)l * IDIM, uc);

    /* ssm = uc @ xp_w^T : M=8192 K=256 N=40 (padded 48) */
    k_gemm<0><<<(MROWS / 16) * 3 / 4, 128, 0, stream>>>(
        uc, xp_w + (size_t)l * 40 * IDIM, ssm, nullptr,
        IDIM, IDIM, IDIM, SSMLD, 3, 40);

    /* dt = softplus(ssm[:, :8] @ dt_w^T + dt_b) : M=8192 K=8 N=256 */
    k_gemm<1><<<(MROWS / 16) * 16 / 4, 128, 0, stream>>>(
        ssm, dt_w + (size_t)l * IDIM * DTRDIM, dtb, dt_b + (size_t)l * IDIM,
        DTRDIM, SSMLD, DTRDIM, IDIM, 16, IDIM);

    k_scan<<<BDIM * (IDIM / 16), 256, 0, stream>>>(
        dtb, uc, ssm, proj, A_log + (size_t)l * IDIM * NDIM, Dp + (size_t)l * IDIM, yb);

    /* x += y @ out_w^T : M=8192 K=256 N=128 (residual in-place) */
    k_gemm<2><<<(MROWS / 16) * 8 / 4, 128, 0, stream>>>(
        yb, out_w + (size_t)l * DDIM * IDIM, x, nullptr,
        IDIM, IDIM, IDIM, DDIM, 8, DDIM);
  }

  k_finalnorm<<<BDIM, 128, 0, stream>>>(x, normf_w, out);
}